// BilateralGridLayer_50036368998638
// MI455X (gfx1250) — compile-verified
//
#include <hip/hip_runtime.h>
#include <hip/hip_bf16.h>

// BilateralGrid: B=8, C=3, H=W=512, G=16, sigma=0.1 -> exp(-50*d^2)
// Outputs (flat, in order): grid_value[128] | wx_full[8*512*512*16]
//                           | wy_full[...] | wz_full[...]

#define GBINS 16
#define HH 512
#define WWID 512
#define BB 8

// exp(-50*d^2) == exp2(d^2 * -50*log2(e)) ; 50*log2(e) = 72.13475204444817
#define NEG50_LOG2E -72.13475204444817f

typedef __attribute__((ext_vector_type(2))) float v2f;
typedef __attribute__((ext_vector_type(8))) float v8f;

__device__ __forceinline__ float fast_exp_n50sq(float d) {
#if defined(__gfx1250__) && __has_builtin(__builtin_amdgcn_exp2f)
    return __builtin_amdgcn_exp2f(d * d * NEG50_LOG2E);   // v_exp_f32
#else
    return __expf(d * d * -50.0f);
#endif
}

__device__ __forceinline__ float fast_rcp(float s) {
#if defined(__gfx1250__) && __has_builtin(__builtin_amdgcn_rcpf)
    return __builtin_amdgcn_rcpf(s);                      // v_rcp_f32
#else
    return 1.0f / s;
#endif
}

// ---------------------------------------------------------------------------
// Kernel 1: spatial soft-bin table. coord_x == coord_y (both linspace(0,1,512))
// so one 512x16 table serves both wx and wy. 8192 floats -> d_ws.
// ---------------------------------------------------------------------------
__global__ void bg_weights_kernel(float* __restrict__ wtab) {
    int c = blockIdx.x * blockDim.x + threadIdx.x;
    if (c >= WWID) return;
    float coord = (float)c * (1.0f / (WWID - 1));
    float w[GBINS];
    float s = 0.0f;
#pragma unroll
    for (int g = 0; g < GBINS; ++g) {
        float d = coord - (float)g * (1.0f / (GBINS - 1));
        w[g] = fast_exp_n50sq(d);
        s += w[g];
    }
    float inv = fast_rcp(s);
#pragma unroll
    for (int g = 0; g < GBINS; ++g) wtab[c * GBINS + g] = w[g] * inv;
}

// ---------------------------------------------------------------------------
// Kernel 2: main streaming kernel. One block per (batch, row): 4096 blocks,
// 256 threads. Thread = (pixel-in-group, bin): g = tid&15. Each wave covers
// 2 pixels x 16 bins -> every b32 NT store is 128B contiguous per wave.
// Per-lane acc of grid_value contributions, reduced per wave via WMMA f32
// 16x16x4 (ones x acc), then fixed-order LDS reduction -> per-block partial.
// ---------------------------------------------------------------------------
__global__ void __launch_bounds__(256)
bg_main_kernel(const float* __restrict__ x, const float* __restrict__ wtab,
               float* __restrict__ wxf, float* __restrict__ wyf,
               float* __restrict__ wzf, float* __restrict__ partials) {
    __shared__ float s_wtab[WWID * GBINS];   // 32 KB of the 320 KB WGP LDS
    __shared__ float s_part[8 * GBINS];

    const int tid = threadIdx.x;
    for (int i = tid; i < WWID * GBINS; i += 256) s_wtab[i] = wtab[i];
    __syncthreads();

    const int g  = tid & 15;        // bin
    const int pl = tid >> 4;        // pixel within 16-pixel pass
    const int h  = blockIdx.x & (HH - 1);
    const int b  = blockIdx.x >> 9;

    const float gridg = (float)g * (1.0f / (GBINS - 1));
    const float wyv   = s_wtab[h * GBINS + g];          // constant per block/lane

    const float* x0 = x + (((size_t)b * 3 + 0) * HH + h) * WWID;
    const float* x1 = x + (((size_t)b * 3 + 1) * HH + h) * WWID;
    const float* x2 = x + (((size_t)b * 3 + 2) * HH + h) * WWID;
    const size_t pix_base = ((size_t)(b * HH + h) * WWID) * GBINS;

    float acc = 0.0f;
#pragma unroll 4
    for (int it = 0; it < WWID / 16; ++it) {
        int w = it * 16 + pl;
        float xv0 = x0[w];
        float xv1 = x1[w];
        float xv2 = x2[w];
        float ww  = s_wtab[w * GBINS + g];

        float e = fast_exp_n50sq(xv2 - gridg);
        // normalize over the 16 bins (16-lane segment reduction)
        float s = e;
        s += __shfl_xor(s, 1, 16);
        s += __shfl_xor(s, 2, 16);
        s += __shfl_xor(s, 4, 16);
        s += __shfl_xor(s, 8, 16);
        float wzv = e * fast_rcp(s);

        size_t o = pix_base + (size_t)w * GBINS + g;
        // 403 MB streamed out >> 192 MB L2: non-temporal stores
        __builtin_nontemporal_store(ww,  wxf + o);
        __builtin_nontemporal_store(wyv, wyf + o);
        __builtin_nontemporal_store(wzv, wzf + o);

        acc = fmaf(ww,  xv0, acc);
        acc = fmaf(wyv, xv1, acc);
        acc = fmaf(wzv, xv2, acc);
    }

    // Per-wave bin reduction: acc lives at column n=lane%16 in both halves.
    const int wave = tid >> 5;
    const int lane = tid & 31;
    float wsum;
#if defined(__gfx1250__) && __has_builtin(__builtin_amdgcn_wmma_f32_16x16x4_f32)
    {
        // A = ones(16x4), B = [acc ; 0]  ->  D[m,n] = acc[n] + acc[n+16]
        v2f a;  a[0] = 1.0f; a[1] = 1.0f;
        v2f bm; bm[0] = acc; bm[1] = 0.0f;
        v8f c = {};
        c = __builtin_amdgcn_wmma_f32_16x16x4_f32(
                /*neg_a=*/false, a, /*neg_b=*/false, bm,
                /*c_mod=*/(short)0, c, /*reuse_a=*/false, /*reuse_b=*/false);
        wsum = c[0];   // every lane: bin-(lane%16) sum for this wave
    }
#else
    wsum = acc + __shfl_xor(acc, 16, 32);
#endif
    if (lane < 16) s_part[wave * GBINS + lane] = wsum;
    __syncthreads();

    if (tid < GBINS) {
        float t = 0.0f;
#pragma unroll
        for (int wv = 0; wv < 8; ++wv) t += s_part[wv * GBINS + tid];
        partials[(size_t)blockIdx.x * GBINS + tid] = t;  // deterministic order
    }
}

// ---------------------------------------------------------------------------
// Kernel 3: deterministic final reduction over the 512 row-blocks per batch.
// ---------------------------------------------------------------------------
__global__ void bg_finalize_kernel(const float* __restrict__ partials,
                                   float* __restrict__ grid_value) {
    int t = threadIdx.x;           // 0..127  -> (b, g)
    if (t >= BB * GBINS) return;
    int b = t >> 4;
    int g = t & 15;
    float s = 0.0f;
    for (int h = 0; h < HH; ++h)
        s += partials[(size_t)(b * HH + h) * GBINS + g];
    grid_value[t] = s;             // shape (8,1,16) flat
}

extern "C" void kernel_launch(void* const* d_in, const int* in_sizes, int n_in,
                              void* d_out, int out_size, void* d_ws, size_t ws_size,
                              hipStream_t stream) {
    (void)in_sizes; (void)n_in; (void)out_size; (void)ws_size;

    const float* x = (const float*)d_in[0];
    float* out = (float*)d_out;

    // workspace: wtab[8192] | partials[4096*16]   (~288 KB)
    float* wtab     = (float*)d_ws;
    float* partials = wtab + WWID * GBINS;

    const size_t full = (size_t)BB * HH * WWID * GBINS;  // 33,554,432
    float* grid_value = out;               // 128 floats
    float* wxf = out + BB * GBINS;         // +128
    float* wyf = wxf + full;
    float* wzf = wyf + full;

    bg_weights_kernel<<<2, 256, 0, stream>>>(wtab);
    bg_main_kernel<<<BB * HH, 256, 0, stream>>>(x, wtab, wxf, wyf, wzf, partials);
    bg_finalize_kernel<<<1, 128, 0, stream>>>(partials, grid_value);
}